// RelevanceFilter_65798898974896
// MI455X (gfx1250) — compile-verified
//
#include <hip/hip_runtime.h>

typedef __attribute__((ext_vector_type(2))) float v2f;
typedef __attribute__((ext_vector_type(8))) float v8f;

#define NB      4
#define CATT    256
#define CVID    512
#define TT      16
#define HH      32
#define WW      32
#define SPATIAL (TT*HH*WW)                         /* 16384 */
#define KV_ELEMS ((size_t)NB * 2 * CATT * SPATIAL) /* 33,554,432 floats */
#define KERN_PER_B 2304                            /* CATT * 1*3*3 */

// ---------------------------------------------------------------------------
// Pass A: kv[b, o, n] = sum_k Wkv_w[o,k] * video[b,k,n] + Wkv_b[o]
// Per-wave 32(M) x 64(N) tile via V_WMMA_F32_16X16X4_F32.
// B fragments (video, the HBM-bound stream) are reused across two M-blocks,
// so the steady-state loop is 2 b64 + 8 b32 loads per 8 WMMAs.
// grid: 2048 blocks x 256 thr = 16384 waves = 4 b * 16 o32-tiles * 256 n64-tiles
// ---------------------------------------------------------------------------
__global__ __launch_bounds__(256) void kv_gemm_wmma(
    const float* __restrict__ Wkv_w,   // (512, 512) row-major
    const float* __restrict__ Wkv_b,   // (512,)
    const float* __restrict__ video,   // (4, 512, 16384)
    float* __restrict__ kv)            // (4, 512, 16384)
{
  const int wave = (blockIdx.x * 256 + threadIdx.x) >> 5;
  const int lane = threadIdx.x & 31;
  const int lo = lane & 15;
  const int hi = lane >> 4;

  const int nt = wave & 255;          // n-tile of 64
  const int ot = (wave >> 8) & 15;    // o-tile of 32
  const int b  = wave >> 12;          // batch
  const int o0 = ot * 32;
  const int n0 = nt * 64;

  // A fragment sources: rows M = lo of the two 16-row M-blocks
  const float* __restrict__ Arow0 = Wkv_w + (size_t)(o0 + lo) * CVID;
  const float* __restrict__ Arow1 = Wkv_w + (size_t)(o0 + 16 + lo) * CVID;
  // B fragment source: column N = n0 + 16*j + lo of video[b]
  const float* __restrict__ Bbase = video + (size_t)b * CVID * SPATIAL + n0 + lo;

  // acc[j]   : M-block 0, N-subtile j
  // acc[4+j] : M-block 1, N-subtile j
  v8f acc0 = {}, acc1 = {}, acc2 = {}, acc3 = {};
  v8f acc4 = {}, acc5 = {}, acc6 = {}, acc7 = {};

  for (int k = 0; k < CVID; k += 4) {
    // A 16x4: VGPR0 = K=(2*hi), VGPR1 = K=(2*hi+1), row M = lo  -> b64 loads
    v2f a0, a1;
    a0.x = Arow0[k + 2 * hi];
    a0.y = Arow0[k + 2 * hi + 1];
    a1.x = Arow1[k + 2 * hi];
    a1.y = Arow1[k + 2 * hi + 1];

    // B 4x16: VGPR0 = row K=(2*hi), VGPR1 = row K=(2*hi+1), col N = lo
    const float* __restrict__ bp = Bbase + (size_t)(k + 2 * hi) * SPATIAL;
    v2f b0, b1, b2, b3;
    b0.x = bp[0];      b0.y = bp[0 + SPATIAL];
    b1.x = bp[16];     b1.y = bp[16 + SPATIAL];
    b2.x = bp[32];     b2.y = bp[32 + SPATIAL];
    b3.x = bp[48];     b3.y = bp[48 + SPATIAL];

    acc0 = __builtin_amdgcn_wmma_f32_16x16x4_f32(false, a0, false, b0, (short)0, acc0, false, false);
    acc1 = __builtin_amdgcn_wmma_f32_16x16x4_f32(false, a0, false, b1, (short)0, acc1, false, false);
    acc2 = __builtin_amdgcn_wmma_f32_16x16x4_f32(false, a0, false, b2, (short)0, acc2, false, false);
    acc3 = __builtin_amdgcn_wmma_f32_16x16x4_f32(false, a0, false, b3, (short)0, acc3, false, false);
    acc4 = __builtin_amdgcn_wmma_f32_16x16x4_f32(false, a1, false, b0, (short)0, acc4, false, false);
    acc5 = __builtin_amdgcn_wmma_f32_16x16x4_f32(false, a1, false, b1, (short)0, acc5, false, false);
    acc6 = __builtin_amdgcn_wmma_f32_16x16x4_f32(false, a1, false, b2, (short)0, acc6, false, false);
    acc7 = __builtin_amdgcn_wmma_f32_16x16x4_f32(false, a1, false, b3, (short)0, acc7, false, false);
  }

  // D layout: VGPR r holds M = r (lanes 0-15) / M = r+8 (lanes 16-31), N = lo
#pragma unroll
  for (int r = 0; r < 8; ++r) {
    const int m = r + 8 * hi;
    {
      const int o = o0 + m;
      const float bias = Wkv_b[o];
      float* __restrict__ row = kv + ((size_t)b * 512 + o) * SPATIAL + n0 + lo;
      row[0]  = acc0[r] + bias;
      row[16] = acc1[r] + bias;
      row[32] = acc2[r] + bias;
      row[48] = acc3[r] + bias;
    }
    {
      const int o = o0 + 16 + m;
      const float bias = Wkv_b[o];
      float* __restrict__ row = kv + ((size_t)b * 512 + o) * SPATIAL + n0 + lo;
      row[0]  = acc4[r] + bias;
      row[16] = acc5[r] + bias;
      row[32] = acc6[r] + bias;
      row[48] = acc7[r] + bias;
    }
  }
}

// ---------------------------------------------------------------------------
// Pass B: kern[b, i] = dot(text_fea[b,:], Wt_w[i,:]) + Wt_b[i]   (4 x 2304)
// ---------------------------------------------------------------------------
__global__ __launch_bounds__(256) void text_kernel_gemm(
    const float* __restrict__ text,    // (4, 512)
    const float* __restrict__ Wt_w,    // (2304, 512)
    const float* __restrict__ Wt_b,    // (2304,)
    float* __restrict__ kern)          // (4, 2304)
{
  const int i = blockIdx.x * 256 + threadIdx.x;
  if (i >= NB * KERN_PER_B) return;
  const int b = i / KERN_PER_B;
  const int r = i % KERN_PER_B;
  const float* __restrict__ t = text + b * CVID;
  const float* __restrict__ w = Wt_w + (size_t)r * CVID;
  float acc = 0.f;
  for (int k = 0; k < CVID; ++k) acc += t[k] * w[k];
  kern[i] = acc + Wt_b[r];
}

// ---------------------------------------------------------------------------
// Pass C: grouped 3x3 conv on k + sigmoid gating, fused with v*sig*mask and
// active_map. One block = (b, t, 8 h-rows); thread = (h_local, w).
// att[b,g,t,h,w] = sum_{c,kh,kw} k[b, g*32+c, t, h+kh-1, w+kw-1] * kern[b,g,c,kh,kw]
// ---------------------------------------------------------------------------
__global__ __launch_bounds__(256) void gate_fused(
    const float* __restrict__ kv,      // (4, 512, 16384); k = ch [0,256), v = [256,512)
    const float* __restrict__ kern,    // (4, 2304)
    const float* __restrict__ masks,   // (4, 1, 16384)
    float* __restrict__ out)           // out (4,256,16384) ++ active (4,1,16384)
{
  const int blk  = blockIdx.x;
  const int hblk = blk & 3;
  const int t    = (blk >> 2) & 15;
  const int b    = blk >> 6;
  const int h    = hblk * 8 + (threadIdx.x >> 5);
  const int w    = threadIdx.x & 31;
  const int thw  = t * (HH * WW) + h * WW + w;

  const float* __restrict__ kbase = kv + (size_t)b * 512 * SPATIAL;
  const float* __restrict__ kernb = kern + b * KERN_PER_B;

  float att[8];
#pragma unroll
  for (int g = 0; g < 8; ++g) {
    float acc = 0.f;
    const float* __restrict__ kg = kernb + g * 288;
    for (int c = 0; c < 32; ++c) {
      const float* __restrict__ kch =
          kbase + (size_t)(g * 32 + c) * SPATIAL + t * (HH * WW);
      const float* __restrict__ kc = kg + c * 9;
#pragma unroll
      for (int kh = 0; kh < 3; ++kh) {
        const int hy = h + kh - 1;
        if (hy < 0 || hy >= HH) continue;
#pragma unroll
        for (int kw = 0; kw < 3; ++kw) {
          const int wx = w + kw - 1;
          if (wx < 0 || wx >= WW) continue;
          acc += kch[hy * WW + wx] * kc[kh * 3 + kw];
        }
      }
    }
    att[g] = acc;
  }

  const float m = masks[b * SPATIAL + thw];

  float sig[8];
  float mean = 0.f;
#pragma unroll
  for (int g = 0; g < 8; ++g) {
    mean += att[g];
    sig[g] = 1.f / (1.f + __expf(-att[g]));
  }
  mean *= 0.125f;
  const float active = m / (1.f + __expf(-mean));

  // active_map concatenated after out
  out[(size_t)NB * CATT * SPATIAL + (size_t)b * SPATIAL + thw] = active;

  const float* __restrict__ vbase = kbase + (size_t)CATT * SPATIAL + thw;
  float* __restrict__ ob = out + (size_t)b * CATT * SPATIAL + thw;
#pragma unroll 4
  for (int ch = 0; ch < CATT; ++ch) {
    const float vv = vbase[(size_t)ch * SPATIAL];
    ob[(size_t)ch * SPATIAL] = vv * sig[ch >> 5] * m;
  }
}

// ---------------------------------------------------------------------------
extern "C" void kernel_launch(void* const* d_in, const int* in_sizes, int n_in,
                              void* d_out, int out_size, void* d_ws, size_t ws_size,
                              hipStream_t stream) {
  const float* video = (const float*)d_in[0];  // (4,512,16,32,32)
  const float* text  = (const float*)d_in[1];  // (4,512)
  const float* masks = (const float*)d_in[2];  // (4,1,16,32,32)
  const float* Wkv_w = (const float*)d_in[3];  // (512,512)
  const float* Wkv_b = (const float*)d_in[4];  // (512,)
  const float* Wt_w  = (const float*)d_in[5];  // (2304,512)
  const float* Wt_b  = (const float*)d_in[6];  // (2304,)
  float* out  = (float*)d_out;

  float* kv   = (float*)d_ws;          // 33,554,432 floats (134 MB)
  float* kern = kv + KV_ELEMS;         // 9,216 floats

  // Pass A: 16384 waves = 2048 blocks x 8 waves
  kv_gemm_wmma<<<2048, 256, 0, stream>>>(Wkv_w, Wkv_b, video, kv);
  // Pass B: 9216 outputs
  text_kernel_gemm<<<(NB * KERN_PER_B + 255) / 256, 256, 0, stream>>>(text, Wt_w, Wt_b, kern);
  // Pass C: 4 b * 16 t * 4 h-blocks = 256 blocks
  gate_fused<<<NB * TT * (HH / 8), 256, 0, stream>>>(kv, kern, masks, out);
}